// ConvCaps_42631845380383
// MI455X (gfx1250) — compile-verified
//
#include <hip/hip_runtime.h>

typedef __attribute__((ext_vector_type(2))) float v2f;
typedef __attribute__((ext_vector_type(8))) float v8f;

#define KDIM      3
#define B_IN_C    32
#define C_OUT_C   32
#define PS_C      16
#define KB_C      288      // 3*3*32
#define OH_C      12
#define OW_C      12
#define NPOS_C    576      // 4*12*12
#define THREADS_C 256
#define EPS_C     1e-6f
#define LAMBDA_C  1e-3f
#define LN_2PI_C  1.8378770664093453f

// LDS plan (bytes):
//   mu   512 f32 = 2048
//   sig  512 f32 = 2048
//   rsum  32 f32 =  128
//   aout  32 f32 =  128
//   ain  288 f32 = 1152
//   v   288*32*16 f16 = 294912
//   r   288*32     f16 =  18432
//   total = 318848  (<= 320KB WGP LDS on CDNA5)
#define SMEM_BYTES ((512 + 512 + 32 + 32 + 288) * 4 + (KB_C * C_OUT_C * PS_C + KB_C * C_OUT_C) * 2)

extern "C" __global__ __launch_bounds__(THREADS_C, 1)
void convcaps_em_kernel(const float* __restrict__ x,   // (4,14,14,32,16)
                        const float* __restrict__ a,   // (4,14,14,32,1)
                        const float* __restrict__ w,   // (288,32,4,4)
                        const float* __restrict__ bu,  // (32,1)
                        const float* __restrict__ ba,  // (32,1)
                        float* __restrict__ out)       // mu (576*512) ++ a_out (576*32)
{
    extern __shared__ char smem[];
    float*    mu   = (float*)smem;                     // 512
    float*    sig  = mu + 512;                         // 512
    float*    rsum = sig + 512;                        // 32
    float*    aout = rsum + 32;                        // 32
    float*    ain  = aout + 32;                        // 288
    _Float16* vl   = (_Float16*)(ain + 288);           // 288*32*16
    _Float16* rl   = vl + KB_C * C_OUT_C * PS_C;       // 288*32

    const int tid  = threadIdx.x;
    const int lane = tid & 31;
    const int wave = tid >> 5;

    const int pos = blockIdx.x;
    const int b   = pos / (OH_C * OW_C);
    const int rem = pos % (OH_C * OW_C);
    const int h0  = rem / OW_C;
    const int w0  = rem % OW_C;

    // ---- stage patch activations a_in[k] ----
    for (int k = tid; k < KB_C; k += THREADS_C) {
        int kh = k / (KDIM * B_IN_C);
        int kw = (k / B_IN_C) % KDIM;
        int bi = k & (B_IN_C - 1);
        ain[k] = a[(((b * 14 + h0 + kh) * 14) + (w0 + kw)) * B_IN_C + bi];
    }

    // ---- pose transform: v[k,c,i,l] = sum_j xp[k,i,j] * W[k,c,j,l] ----
    // One V_WMMA_F32_16X16X4_F32 per (k, 4-output-cap chunk).
    // A (16x4 f32): lanes carry M (rows 0..3 = i, rest zero-padded);
    //               VGPR0/1 carry K split {0,1}(lanes<16) / {2,3}(lanes>=16).
    // B (4x16 f32): lanes carry N = (c_sub*4 + l); same K split per VGPR.
    {
        const int m    = lane & 15;
        const int jlo  = (lane < 16) ? 0 : 2;
        const int n    = lane & 15;
        const int csub = n >> 2;
        const int l    = n & 3;
        for (int k = wave; k < KB_C; k += 8) {          // uniform per wave: EXEC all-1s at WMMA
            int kh = k / (KDIM * B_IN_C);
            int kw = (k / B_IN_C) % KDIM;
            int bi = k & (B_IN_C - 1);
            const float* xp = x + ((((b * 14 + h0 + kh) * 14) + (w0 + kw)) * B_IN_C + bi) * PS_C;
            v2f A;
            if (m < 4) { A.x = xp[m * 4 + jlo]; A.y = xp[m * 4 + jlo + 1]; }
            else       { A.x = 0.0f;            A.y = 0.0f; }
            #pragma unroll
            for (int cb = 0; cb < C_OUT_C; cb += 4) {
                const float* wp = w + ((k * C_OUT_C + cb + csub) * 4 + jlo) * 4 + l;
                v2f Bf; Bf.x = wp[0]; Bf.y = wp[4];     // j and j+1 (stride 4 floats)
                v8f C = {};
                C = __builtin_amdgcn_wmma_f32_16x16x4_f32(
                        /*neg_a=*/false, A, /*neg_b=*/false, Bf,
                        /*c_mod=*/(short)0, C, /*reuse_a=*/false, /*reuse_b=*/false);
                if (lane < 16) {                         // D: VGPR r <-> row i=r (lanes 0..15)
                    _Float16* vp = vl + (k * C_OUT_C + cb + csub) * PS_C + l;
                    vp[0]  = (_Float16)C[0];
                    vp[4]  = (_Float16)C[1];
                    vp[8]  = (_Float16)C[2];
                    vp[12] = (_Float16)C[3];
                }
            }
        }
    }
    __syncthreads();

    // ---- EM routing, 3 iterations, all state in LDS ----
    for (int it = 0; it < 3; ++it) {
        // r[k,c]
        if (it == 0) {
            for (int k = tid; k < KB_C; k += THREADS_C) {
                _Float16 rv = (_Float16)(ain[k] * (1.0f / C_OUT_C));
                #pragma unroll
                for (int c = 0; c < C_OUT_C; ++c) rl[k * C_OUT_C + c] = rv;
            }
        } else {
            for (int k = tid; k < KB_C; k += THREADS_C) {
                float lp[C_OUT_C];
                float mx = -1e30f;
                #pragma unroll
                for (int c = 0; c < C_OUT_C; ++c) {
                    const _Float16* vp = vl + (k * C_OUT_C + c) * PS_C;
                    float s = 0.0f;
                    for (int p = 0; p < PS_C; ++p) {
                        float vv = (float)vp[p];
                        float d  = vv - mu[c * PS_C + p];
                        float sg = sig[c * PS_C + p];
                        s += -d * d / (2.0f * sg) - 0.5f * __logf(sg);
                    }
                    s += -8.0f * LN_2PI_C + __logf(aout[c]);
                    lp[c] = s;
                    mx = fmaxf(mx, s);
                }
                float se = 0.0f;
                #pragma unroll
                for (int c = 0; c < C_OUT_C; ++c) { lp[c] = __expf(lp[c] - mx); se += lp[c]; }
                float inv = 1.0f / se;
                #pragma unroll
                for (int c = 0; c < C_OUT_C; ++c)
                    rl[k * C_OUT_C + c] = (_Float16)(lp[c] * inv * aout[c]);
            }
        }
        __syncthreads();

        // r_sum[c] = sum_k r[k,c]
        if (tid < C_OUT_C) {
            float s = 0.0f;
            for (int k = 0; k < KB_C; ++k) s += (float)rl[k * C_OUT_C + tid];
            rsum[tid] = s;
        }
        __syncthreads();

        // mu[c,p] = sum_k r*v / (r_sum + eps)
        for (int idx = tid; idx < C_OUT_C * PS_C; idx += THREADS_C) {
            int c = idx >> 4, p = idx & 15;
            float acc = 0.0f;
            for (int k = 0; k < KB_C; ++k)
                acc += (float)rl[k * C_OUT_C + c] * (float)vl[(k * C_OUT_C + c) * PS_C + p];
            mu[idx] = acc / (rsum[c] + EPS_C);
        }
        __syncthreads();

        // sigma[c,p] = sum_k r*(v-mu)^2 / (r_sum + eps) + eps
        for (int idx = tid; idx < C_OUT_C * PS_C; idx += THREADS_C) {
            int c = idx >> 4, p = idx & 15;
            float mm  = mu[idx];
            float acc = 0.0f;
            for (int k = 0; k < KB_C; ++k) {
                float dv = (float)vl[(k * C_OUT_C + c) * PS_C + p] - mm;
                acc += (float)rl[k * C_OUT_C + c] * dv * dv;
            }
            sig[idx] = acc / (rsum[c] + EPS_C) + EPS_C;
        }
        __syncthreads();

        // a_out[c] = sigmoid(lambda*(beta_a - r_sum*sum_p(beta_u + 0.5 log sigma)))
        if (tid < C_OUT_C) {
            float s = 0.0f;
            for (int p = 0; p < PS_C; ++p) s += bu[tid] + 0.5f * __logf(sig[tid * PS_C + p]);
            float cost = s * rsum[tid];
            aout[tid] = 1.0f / (1.0f + __expf(-(LAMBDA_C * (ba[tid] - cost))));
        }
        __syncthreads();
    }

    // ---- outputs: mu then a_out, concatenated flat ----
    for (int idx = tid; idx < C_OUT_C * PS_C; idx += THREADS_C)
        out[pos * (C_OUT_C * PS_C) + idx] = mu[idx];
    if (tid < C_OUT_C)
        out[NPOS_C * (C_OUT_C * PS_C) + pos * C_OUT_C + tid] = aout[tid];
}

extern "C" void kernel_launch(void* const* d_in, const int* in_sizes, int n_in,
                              void* d_out, int out_size, void* d_ws, size_t ws_size,
                              hipStream_t stream) {
    (void)in_sizes; (void)n_in; (void)d_ws; (void)ws_size; (void)out_size;
    const float* x  = (const float*)d_in[0];
    const float* a  = (const float*)d_in[1];
    const float* w  = (const float*)d_in[2];
    const float* bu = (const float*)d_in[3];
    const float* ba = (const float*)d_in[4];
    float* out = (float*)d_out;
    convcaps_em_kernel<<<NPOS_C, THREADS_C, SMEM_BYTES, stream>>>(x, a, w, bu, ba, out);
}